// Encoder_90915867722224
// MI455X (gfx1250) — compile-verified
//
#include <hip/hip_runtime.h>
#include <math.h>

typedef float v2f __attribute__((ext_vector_type(2)));
typedef float v8f __attribute__((ext_vector_type(8)));

// ---------------------------------------------------------------------------
// Dense GEMM + bias via V_WMMA_F32_16X16X4_F32.
// Y[N,DOUT] = X[N,128] @ W[128,DOUT] + bias.
// Block: 16 rows, (DOUT/16) waves; wave w owns column tile [16w, 16w+16).
// ---------------------------------------------------------------------------
template <int DOUT>
__global__ void gemm_bias_wmma(const float* __restrict__ X,
                               const float* __restrict__ W,
                               const float* __restrict__ bias,
                               float* __restrict__ Y, int N) {
  constexpr int K = 128;
  const int lane = threadIdx.x & 31;
  const int wave = threadIdx.x >> 5;
  const int half = lane >> 4;   // 0: lanes 0-15, 1: lanes 16-31
  const int lr   = lane & 15;
  const int row0 = blockIdx.x * 16;
  const int col0 = wave * 16;

  // Clamp read row so EXEC stays all-ones for WMMA (tail rows masked at store).
  int arow = row0 + lr;
  if (arow >= N) arow = N - 1;
  const float* xrow = X + (size_t)arow * K;

  v8f acc = {};
#pragma unroll
  for (int kk = 0; kk < K; kk += 4) {
    const int k = kk + half * 2;  // ISA A-layout: lanes 0-15 hold K+0,K+1; lanes 16-31 hold K+2,K+3
    v2f a;
    a.x = xrow[k];
    a.y = xrow[k + 1];
    v2f b;  // B-layout mirrors A with N striped across lanes
    b.x = W[(size_t)k * DOUT + col0 + lr];
    b.y = W[(size_t)(k + 1) * DOUT + col0 + lr];
    acc = __builtin_amdgcn_wmma_f32_16x16x4_f32(
        /*neg_a=*/false, a, /*neg_b=*/false, b,
        /*c_mod=*/(short)0, acc, /*reuse_a=*/false, /*reuse_b=*/false);
  }

  const float bc = bias[col0 + lr];
#pragma unroll
  for (int v = 0; v < 8; ++v) {
    // C/D layout: lanes 0-15 -> M = v, lanes 16-31 -> M = v + 8; N = lane&15
    const int row = row0 + v + half * 8;
    if (row < N) Y[(size_t)row * DOUT + col0 + lr] = acc[v] + bc;
  }
}

// ---------------------------------------------------------------------------
// Degree: deg[n] = (# edges with row==n) + 1 ;  dinv = deg^-0.5
// ---------------------------------------------------------------------------
__global__ void deg_count(const long long* __restrict__ ei, float* __restrict__ deg, int E) {
  int e = blockIdx.x * blockDim.x + threadIdx.x;
  if (e < E) atomicAdd(&deg[(int)ei[e]], 1.0f);
}

__global__ void deg_finalize(float* __restrict__ deg, float* __restrict__ dinv, int N) {
  int n = blockIdx.x * blockDim.x + threadIdx.x;
  if (n < N) {
    float d = deg[n] + 1.0f;
    deg[n]  = d;
    dinv[n] = 1.0f / sqrtf(d);
  }
}

// ---------------------------------------------------------------------------
// Edge message + scatter:
//   agg[col] += dinv[row]*dinv[col] * relu( H[row] + edge_attr[e] @ We + be )
// DOUT/4 threads per edge, float4 per thread; We,be cached in LDS;
// e-vector recomputed on the fly (7 FMAs per float4 component).
// ---------------------------------------------------------------------------
template <int DOUT>
__global__ void edge_scatter(const long long* __restrict__ ei,
                             const float* __restrict__ eattr,
                             const float* __restrict__ We,
                             const float* __restrict__ be,
                             const float* __restrict__ H,
                             const float* __restrict__ dinv,
                             float* __restrict__ agg, int E) {
  constexpr int TPE = DOUT / 4;   // threads per edge (32 or 16)
  constexpr int EPB = 256 / TPE;  // edges per 256-thread block
  __shared__ float sWe[7 * DOUT];
  __shared__ float sbe[DOUT];
  for (int i = threadIdx.x; i < 7 * DOUT; i += 256) sWe[i] = We[i];
  for (int i = threadIdx.x; i < DOUT; i += 256) sbe[i] = be[i];
  __syncthreads();

  const int eid = blockIdx.x * EPB + threadIdx.x / TPE;
  if (eid >= E) return;
  const int c0 = (threadIdx.x % TPE) * 4;

  const int r = (int)ei[eid];
  const int c = (int)ei[(size_t)E + eid];
  const float nrm = dinv[r] * dinv[c];

  float ea[7];
#pragma unroll
  for (int j = 0; j < 7; ++j) ea[j] = eattr[(size_t)eid * 7 + j];

  float4 ev = *(const float4*)(sbe + c0);
#pragma unroll
  for (int j = 0; j < 7; ++j) {
    const float4 w = *(const float4*)(sWe + j * DOUT + c0);
    ev.x += ea[j] * w.x;
    ev.y += ea[j] * w.y;
    ev.z += ea[j] * w.z;
    ev.w += ea[j] * w.w;
  }

  const float4 h = *(const float4*)(H + (size_t)r * DOUT + c0);
  const float m0 = fmaxf(h.x + ev.x, 0.0f) * nrm;
  const float m1 = fmaxf(h.y + ev.y, 0.0f) * nrm;
  const float m2 = fmaxf(h.z + ev.z, 0.0f) * nrm;
  const float m3 = fmaxf(h.w + ev.w, 0.0f) * nrm;

  float* ap = agg + (size_t)c * DOUT + c0;
  atomicAdd(ap + 0, m0);
  atomicAdd(ap + 1, m1);
  atomicAdd(ap + 2, m2);
  atomicAdd(ap + 3, m3);
}

// ---------------------------------------------------------------------------
// Node combine: out = [relu]( agg + relu(H + root)/deg )
// ---------------------------------------------------------------------------
template <int DOUT, bool RELU>
__global__ void node_combine(const float* __restrict__ agg,
                             const float* __restrict__ H,
                             const float* __restrict__ root,
                             const float* __restrict__ deg,
                             float* __restrict__ out, int N) {
  const size_t idx = (size_t)blockIdx.x * blockDim.x + threadIdx.x;
  if (idx >= (size_t)N * DOUT) return;
  const int c = (int)(idx % DOUT);
  const size_t n = idx / DOUT;
  float v = agg[idx] + fmaxf(H[idx] + root[c], 0.0f) / deg[n];
  out[idx] = RELU ? fmaxf(v, 0.0f) : v;
}

// ---------------------------------------------------------------------------
extern "C" void kernel_launch(void* const* d_in, const int* in_sizes, int n_in,
                              void* d_out, int out_size, void* d_ws, size_t ws_size,
                              hipStream_t stream) {
  const float*     x      = (const float*)d_in[0];
  const long long* ei     = (const long long*)d_in[1];
  const float*     eattr  = (const float*)d_in[2];
  const float*     W1     = (const float*)d_in[3];
  const float*     b1     = (const float*)d_in[4];
  const float*     We1    = (const float*)d_in[5];
  const float*     be1    = (const float*)d_in[6];
  const float*     root1  = (const float*)d_in[7];
  const float*     Wmu    = (const float*)d_in[8];
  const float*     bmu    = (const float*)d_in[9];
  const float*     Wemu   = (const float*)d_in[10];
  const float*     bemu   = (const float*)d_in[11];
  const float*     rootmu = (const float*)d_in[12];
  const float*     Wls    = (const float*)d_in[13];
  const float*     bls    = (const float*)d_in[14];
  const float*     Wels   = (const float*)d_in[15];
  const float*     bels   = (const float*)d_in[16];
  const float*     rootls = (const float*)d_in[17];

  const int N = in_sizes[0] / 128;
  const int E = in_sizes[1] / 2;

  // Workspace layout (aliased by lifetime):
  //   deg[N], dinv[N], bufA[N*128], bufB[N*128], bufC[N*128]
  float* ws   = (float*)d_ws;
  float* deg  = ws;
  float* dinv = ws + N;
  float* bufA = ws + 2 * (size_t)N;          // Hlin1, later aggmu|aggls
  float* bufB = bufA + (size_t)N * 128;      // agg1,  later Hmu|Hls
  float* bufC = bufB + (size_t)N * 128;      // h1

  float* Hlin1 = bufA;
  float* agg1  = bufB;
  float* h1    = bufC;
  float* Hmu   = bufB;
  float* Hls   = bufB + (size_t)N * 64;
  float* aggmu = bufA;
  float* aggls = bufA + (size_t)N * 64;
  float* outmu = (float*)d_out;
  float* outls = (float*)d_out + (size_t)N * 64;

  // Degree / normalization (shared by all three convs).
  hipMemsetAsync(deg, 0, (size_t)N * sizeof(float), stream);
  deg_count<<<(E + 255) / 256, 256, 0, stream>>>(ei, deg, E);
  deg_finalize<<<(N + 255) / 256, 256, 0, stream>>>(deg, dinv, N);

  // ---- Layer 1 (dout=128) ----
  gemm_bias_wmma<128><<<(N + 15) / 16, 256, 0, stream>>>(x, W1, b1, Hlin1, N);
  hipMemsetAsync(agg1, 0, (size_t)N * 128 * sizeof(float), stream);
  edge_scatter<128><<<(E + 7) / 8, 256, 0, stream>>>(ei, eattr, We1, be1, Hlin1, dinv, agg1, E);
  node_combine<128, true><<<(int)(((size_t)N * 128 + 255) / 256), 256, 0, stream>>>(
      agg1, Hlin1, root1, deg, h1, N);

  // ---- mu / logstd GEMMs (dout=64) ----
  gemm_bias_wmma<64><<<(N + 15) / 16, 128, 0, stream>>>(h1, Wmu, bmu, Hmu, N);
  gemm_bias_wmma<64><<<(N + 15) / 16, 128, 0, stream>>>(h1, Wls, bls, Hls, N);

  // ---- mu / logstd edge scatter + combine ----
  hipMemsetAsync(bufA, 0, (size_t)N * 128 * sizeof(float), stream);  // aggmu|aggls
  edge_scatter<64><<<(E + 15) / 16, 256, 0, stream>>>(ei, eattr, Wemu, bemu, Hmu, dinv, aggmu, E);
  edge_scatter<64><<<(E + 15) / 16, 256, 0, stream>>>(ei, eattr, Wels, bels, Hls, dinv, aggls, E);
  node_combine<64, false><<<(int)(((size_t)N * 64 + 255) / 256), 256, 0, stream>>>(
      aggmu, Hmu, rootmu, deg, outmu, N);
  node_combine<64, false><<<(int)(((size_t)N * 64 + 255) / 256), 256, 0, stream>>>(
      aggls, Hls, rootls, deg, outls, N);
}